// MyContraLoss_71880572665957
// MI455X (gfx1250) — compile-verified
//
#include <hip/hip_runtime.h>
#include <hip/hip_bf16.h>

typedef float v2f __attribute__((ext_vector_type(2)));
typedef float v8f __attribute__((ext_vector_type(8)));

// Problem shape (fixed by the reference harness)
constexpr int Bc = 16;
constexpr int Tc = 2048;
constexpr int Dc = 256;
constexpr int Nc = 128;
constexpr int Kc = 100;
constexpr int BT = Bc * Tc;   // 32768 negatives
constexpr int BN = Bc * Nc;   // 2048 anchors
constexpr int A_STRIDE = 260; // padded LDS row stride (floats) -> conflict-free b64 reads
constexpr float TEMP_INV = 10.0f; // 1/0.1

// ---------------------------------------------------------------------------
// order-preserving float -> uint key (ascending)
__device__ __forceinline__ unsigned orderKey(float f) {
  unsigned u = __float_as_uint(f);
  return (u & 0x80000000u) ? ~u : (u | 0x80000000u);
}
__device__ __forceinline__ float keyToFloat(unsigned k) {
  unsigned u = (k & 0x80000000u) ? (k & 0x7FFFFFFFu) : ~k;
  return __uint_as_float(u);
}

// 16x16 tile of anchors(16xD) . negatives(colsxD)^T via fp32 WMMA, K=256 in 64 steps
__device__ __forceinline__ v8f tile_gemm(const float* __restrict__ ldsA,
                                         const float* __restrict__ tgt,
                                         int col0, int lane) {
  const int half = lane >> 4;     // K sub-pair select per documented fp32 A/B layout
  const int mr   = lane & 15;     // row (A) / column (B,C) index within tile
  const float* arow = ldsA + mr * A_STRIDE + half * 2;
  const float* brow = tgt + ((size_t)(col0 + mr)) * Dc + half * 2;
  v8f acc = {0.f, 0.f, 0.f, 0.f, 0.f, 0.f, 0.f, 0.f};
#pragma unroll 8
  for (int kk = 0; kk < Dc / 4; ++kk) {
    v2f a = *(const v2f*)(arow + kk * 4);
    v2f b = *(const v2f*)(brow + kk * 4);
    acc = __builtin_amdgcn_wmma_f32_16x16x4_f32(false, a, false, b,
                                                (short)0, acc, false, false);
  }
  return acc;
}

// ---------------------------------------------------------------------------
// per-row inverse L2 norms of target rows (one wave per row).
// Full-precision 1/sqrt once per row; hot loop then only multiplies.
__global__ __launch_bounds__(256) void rownorm_kernel(const float* __restrict__ x,
                                                      float* __restrict__ inv_out) {
  int wave = (blockIdx.x * blockDim.x + threadIdx.x) >> 5;
  int lane = threadIdx.x & 31;
  if (wave >= BT) return;
  const float* row = x + (size_t)wave * Dc;
  float s = 0.f;
#pragma unroll
  for (int i = 0; i < Dc / 32; ++i) {
    float v = row[lane + i * 32];
    s += v * v;
  }
#pragma unroll
  for (int off = 16; off >= 1; off >>= 1) s += __shfl_xor(s, off, 32);
  if (lane == 0) inv_out[wave] = 1.0f / sqrtf(s);
}

// anchor inverse norms + numerator acos/T (one wave per anchor row)
__global__ __launch_bounds__(256) void anchor_kernel(const float* __restrict__ ctx,
                                                     const float* __restrict__ tgt,
                                                     const int* __restrict__ midx,
                                                     float* __restrict__ a_inv_norm,
                                                     float* __restrict__ num_acos) {
  int wave = (blockIdx.x * blockDim.x + threadIdx.x) >> 5;
  int lane = threadIdx.x & 31;
  if (wave >= BN) return;
  int b = wave / Nc, n = wave % Nc;
  int t = midx[n];
  const float* a = ctx + ((size_t)b * Tc + t) * Dc;
  const float* p = tgt + ((size_t)b * Tc + t) * Dc;
  float saa = 0.f, spp = 0.f, sap = 0.f;
#pragma unroll
  for (int i = 0; i < Dc / 32; ++i) {
    float av = a[lane + i * 32], pv = p[lane + i * 32];
    saa += av * av;
    spp += pv * pv;
    sap += av * pv;
  }
#pragma unroll
  for (int off = 16; off >= 1; off >>= 1) {
    saa += __shfl_xor(saa, off, 32);
    spp += __shfl_xor(spp, off, 32);
    sap += __shfl_xor(sap, off, 32);
  }
  if (lane == 0) {
    float an = sqrtf(saa), pn = sqrtf(spp);
    a_inv_norm[wave] = 1.0f / an;
    float c = sap / (an * pn);
    c = fminf(1.0f, fmaxf(-1.0f, c));
    num_acos[wave] = acosf(c) * TEMP_INV; // log(sim_numerator)
  }
}

// ---------------------------------------------------------------------------
// main fused kernel: 16 anchor rows per workgroup, 8 waves x 16 cols per sweep.
// 4 radix-select sweeps (exact 100th-smallest cosine key) + 1 summation sweep.
__global__ __launch_bounds__(256) void contra_main(const float* __restrict__ ctx,
                                                   const float* __restrict__ tgt,
                                                   const int* __restrict__ midx,
                                                   const int* __restrict__ phone,
                                                   const float* __restrict__ n_inv,
                                                   const float* __restrict__ a_inv,
                                                   const float* __restrict__ num_acos,
                                                   float* __restrict__ row_logit) {
  __shared__ float    ldsA[16 * A_STRIDE];
  __shared__ float    s_inv_an[16];
  __shared__ int      s_ph[16];
  __shared__ int      s_mi[16];
  __shared__ unsigned s_hist[16][256];
  __shared__ unsigned s_pref[16];
  __shared__ unsigned s_krem[16];
  __shared__ float    s_part[16][8];

  const int tid = threadIdx.x;
  const int r0 = blockIdx.x * 16;   // anchor rows [r0, r0+16)
  const int b  = r0 / Nc;
  const int n0 = r0 % Nc;

  if (tid < 16) s_mi[tid] = midx[n0 + tid];
  __syncthreads();
#pragma unroll 1
  for (int m = 0; m < 16; ++m) { // blockDim == D: one float per thread per row
    ldsA[m * A_STRIDE + tid] = ctx[((size_t)b * Tc + s_mi[m]) * Dc + tid];
  }
  if (tid < 16) {
    s_inv_an[tid] = a_inv[r0 + tid];
    s_ph[tid] = phone[b * Tc + s_mi[tid]];
    s_pref[tid] = 0u;
    s_krem[tid] = (unsigned)Kc;
  }
  __syncthreads();

  const int wave = tid >> 5, lane = tid & 31;
  const int half = lane >> 4, mr = lane & 15;
  constexpr int ITERS = BT / (16 * 8); // 256 sweeps of 128 columns

  // ---- 4 radix-select histogram sweeps (8 key bits each) ----
  for (int pass = 0; pass < 4; ++pass) {
    for (int i = tid; i < 16 * 256; i += 256) ((unsigned*)s_hist)[i] = 0u;
    __syncthreads();
    const int shift = 24 - 8 * pass;
    unsigned hmask = 0u; // resolved-prefix mask
    if (pass == 1) hmask = 0xFF000000u;
    else if (pass == 2) hmask = 0xFFFF0000u;
    else if (pass == 3) hmask = 0xFFFFFF00u;

    for (int iter = 0; iter < ITERS; ++iter) {
      const int col0 = iter * 128 + wave * 16;
      v8f acc = tile_gemm(ldsA, tgt, col0, lane);
      const int j = col0 + mr;            // flat column in [0, B*T)
      const float inv_nn = n_inv[j];      // precomputed reciprocal: no div in hot loop
      const int pt = phone[j];
#pragma unroll
      for (int v = 0; v < 8; ++v) {
        const int m = v + 8 * half;
        float c = acc[v] * s_inv_an[m] * inv_nn;
        c = fminf(1.0f, fmaxf(-1.0f, c));
        unsigned key = (s_ph[m] != pt) ? orderKey(c) : 0xFFFFFFFFu;
        if (((key ^ s_pref[m]) & hmask) == 0u)
          atomicAdd(&s_hist[m][(key >> shift) & 0xFF], 1u);
      }
    }
    __syncthreads();
    if (tid < 16) { // scan 256 bins ascending; refine prefix + residual rank
      unsigned kr = s_krem[tid], cum = 0u, sel = 255u;
      for (int bin = 0; bin < 256; ++bin) {
        unsigned cv = s_hist[tid][bin];
        if (cum + cv >= kr) { sel = (unsigned)bin; break; }
        cum += cv;
      }
      s_krem[tid] = kr - cum;
      s_pref[tid] |= sel << shift;
    }
    __syncthreads();
  }

  // ---- summation sweep: sum sims strictly below threshold key ----
  float accum[8];
#pragma unroll
  for (int v = 0; v < 8; ++v) accum[v] = 0.f;
  for (int iter = 0; iter < ITERS; ++iter) {
    const int col0 = iter * 128 + wave * 16;
    v8f acc = tile_gemm(ldsA, tgt, col0, lane);
    const int j = col0 + mr;
    const float inv_nn = n_inv[j];
    const int pt = phone[j];
#pragma unroll
    for (int v = 0; v < 8; ++v) {
      const int m = v + 8 * half;
      float c = acc[v] * s_inv_an[m] * inv_nn;
      c = fminf(1.0f, fmaxf(-1.0f, c));
      if (s_ph[m] != pt) {
        unsigned key = orderKey(c);
        if (key < s_pref[m]) accum[v] += expf(acosf(c) * TEMP_INV);
      }
    }
  }
  // fixed-order reduction across the 16 lanes sharing each row half
#pragma unroll
  for (int v = 0; v < 8; ++v) {
    float s = accum[v];
#pragma unroll
    for (int off = 8; off >= 1; off >>= 1) s += __shfl_xor(s, off, 16);
    if (mr == 0) s_part[v + 8 * half][wave] = s;
  }
  __syncthreads();
  if (tid < 16) {
    float den = 0.f;
#pragma unroll
    for (int w = 0; w < 8; ++w) den += s_part[tid][w]; // fixed wave order
    const unsigned thr = s_pref[tid];
    float ft = keyToFloat(thr);
    ft = fminf(1.0f, fmaxf(-1.0f, ft));
    den += (float)s_krem[tid] * expf(acosf(ft) * TEMP_INV); // exact tie handling
    row_logit[r0 + tid] = num_acos[r0 + tid] - logf(den);
  }
}

// ---------------------------------------------------------------------------
__global__ __launch_bounds__(256) void final_reduce(const float* __restrict__ row_logit,
                                                    float* __restrict__ out) {
  __shared__ float sm[256];
  int tid = threadIdx.x;
  float s = 0.f;
  for (int i = tid; i < BN; i += 256) s += row_logit[i];
  sm[tid] = s;
  __syncthreads();
  for (int st = 128; st >= 1; st >>= 1) {
    if (tid < st) sm[tid] += sm[tid + st];
    __syncthreads();
  }
  if (tid == 0) out[0] = -sm[0] / (float)BN;
}

// ---------------------------------------------------------------------------
extern "C" void kernel_launch(void* const* d_in, const int* in_sizes, int n_in,
                              void* d_out, int out_size, void* d_ws, size_t ws_size,
                              hipStream_t stream) {
  (void)in_sizes; (void)n_in; (void)out_size; (void)ws_size;
  const float* ctx   = (const float*)d_in[0]; // context_feature (B,T,D)
  const float* tgt   = (const float*)d_in[1]; // target_feature  (B,T,D)
  const int*   midx  = (const int*)d_in[2];   // mask_index (N)
  const int*   phone = (const int*)d_in[3];   // phone_label (B,T)
  float* out = (float*)d_out;

  float* ws      = (float*)d_ws;
  float* n_inv   = ws;                 // BT floats (1 / ||negative_j||)
  float* a_inv   = ws + BT;            // BN floats (1 / ||anchor_r||)
  float* num_ac  = ws + BT + BN;       // BN floats (acos(cos_pos)/T)
  float* logits  = ws + BT + 2 * BN;   // BN floats

  rownorm_kernel<<<BT / 8, 256, 0, stream>>>(tgt, n_inv);
  anchor_kernel<<<BN / 8, 256, 0, stream>>>(ctx, tgt, midx, a_inv, num_ac);
  contra_main<<<BN / 16, 256, 0, stream>>>(ctx, tgt, midx, phone,
                                           n_inv, a_inv, num_ac, logits);
  final_reduce<<<1, 256, 0, stream>>>(logits, out);
}